// PSP_21517786153623
// MI455X (gfx1250) — compile-verified
//
#include <hip/hip_runtime.h>

typedef float v2f __attribute__((ext_vector_type(2)));
typedef float v4f __attribute__((ext_vector_type(4)));
typedef float v8f __attribute__((ext_vector_type(8)));
typedef unsigned int u32;
typedef unsigned long long u64;
typedef u32 v4u __attribute__((ext_vector_type(4)));
typedef int  v4i __attribute__((ext_vector_type(4)));
typedef int  v8i __attribute__((ext_vector_type(8)));

#define NB 64
#define TT 512
#define DD 256
#define BT_ELEMS (NB * TT * DD)      // 8388608
#define ATT_ELEMS (NB * TT * TT)     // 16777216

#define KB    32     // K chunk staged in LDS
#define PADA  36     // A LDS row stride: 32 data + 4 pad (conflict-free frag reads)
#define PADBT 36     // B LDS row stride, transposed-B case (K-packed)
#define PADBN 68     // B LDS row stride, row-major-B case (64 data + 4 pad)

#if defined(__has_builtin)
#if __has_builtin(__builtin_amdgcn_tensor_load_to_lds)
#define USE_TDM 1
#endif
#endif

#ifdef USE_TDM
// Raw LDS byte offset of a __shared__ pointer (generic -> addrspace(3)).
__device__ __forceinline__ u32 lds_off(const void* p)
{
    return (u32)(u64)(const __attribute__((address_space(3))) void*)p;
}

// Issue one TENSOR_LOAD_TO_LDS for a 2-D tile:
//   global: tile_d1 rows of tile_d0 contiguous f32, row stride = stride0 elems
//   LDS:    rows padded by (pad_amt) DWORDs every (pad_int) row
// D# group0/group1 packed per CDNA5 ISA 8.3/8.4; groups 2/3 zero (2-D tensor).
// This toolchain exposes the 6-arg builtin:
//   (uint32x4 g0, int32x8 g1, int32x4 g2, int32x4 g3, int32x8, i32 cpol)
__device__ __forceinline__ void tdm_load_2d(const float* gptr, u32 lds_byte,
                                            u32 tile_d0, u32 tile_d1,
                                            u32 stride0_elems,
                                            u32 pad_int_code, u32 pad_amt_code)
{
    const u64 ga = (u64)gptr;
    v4u g0;
    g0.x = 1u;                                              // count=1, user desc
    g0.y = lds_byte;                                        // lds_addr
    g0.z = (u32)ga;                                         // global_addr[31:0]
    g0.w = (u32)((ga >> 32) & 0x1FFFFFFull) | (2u << 30);   // addr[56:32], type=2
    const u32 td0 = 0x40000000u;   // generous tensor dims -> no OOB clipping
    const u32 td1 = 0x00100000u;
    v8i g1;
    g1[0] = (int)((2u << 16) | (1u << 20) |                 // data_size=4B, pad_en
                  (pad_int_code << 22) | (pad_amt_code << 25));
    g1[1] = (int)((td0 & 0xFFFFu) << 16);                   // tensor_dim0[15:0]
    g1[2] = (int)((td0 >> 16) | ((td1 & 0xFFFFu) << 16));   // dim0[31:16]|dim1[15:0]
    g1[3] = (int)((td1 >> 16) | (tile_d0 << 16));           // dim1[31:16]|tile_dim0
    g1[4] = (int)tile_d1;                                   // tile_dim1, tile_dim2=0
    g1[5] = (int)stride0_elems;                             // tensor_dim0_stride lo
    g1[6] = 0;
    g1[7] = 0;
    __builtin_amdgcn_tensor_load_to_lds(g0, g1, (v4i)0, (v4i)0, (v8i)0, 0);
}
#endif

// ---------------------------------------------------------------------------
// Double-buffered LDS-tiled GEMM on v_wmma_f32_16x16x4_f32 (wave32).
//   C[b] = epilogue(A[b] @ B[b])      A:[M,K] row-major
//   BT=false: B:[K,N] row-major       BT=true: B arg is Bt:[N,K] (C = A@Bt^T)
// Block = 128 threads = 4 waves; block tile 128x64; wave tile 32x64.
// Staging: TDM (TENSOR_LOAD_TO_LDS, TENSORcnt) when available; else regs.
// grid = (M/128, N/64, batch)
// ---------------------------------------------------------------------------
template <bool BT>
__global__ __launch_bounds__(128)
void gemm_tiled(const float* __restrict__ A, const float* __restrict__ B,
                float* __restrict__ C, int K, int N,
                long sA, long sB, long sC, float scale, int do_relu)
{
    __shared__ float Al[2][128 * PADA];   // 36 KB
    __shared__ float Bl[2][2304];         // 18 KB (covers 64*36 and 32*68)

    const int t = threadIdx.x;
    const int lane = t & 31;
    const int w = t >> 5;
    const int l = lane & 15;
    const int h = lane >> 4;

    const float* Ab = A + (size_t)blockIdx.z * sA;
    const float* Bb = B + (size_t)blockIdx.z * sB;
    float*       Cb = C + (size_t)blockIdx.z * sC;

    const int mblk = blockIdx.x * 128;
    const int n0   = blockIdx.y * 64;

    v8f acc[2][4] = {};
    const int nchunks = K / KB;

#ifdef USE_TDM
    // ---- chunk 0: one TDM descriptor per operand (wave0 = A, wave1 = B) ----
    if (w == 0)
        tdm_load_2d(Ab + (size_t)mblk * K, lds_off(&Al[0][0]),
                    KB, 128, (u32)K, 4, 3);
    if (w == 1) {
        if (BT) tdm_load_2d(Bb + (size_t)n0 * K, lds_off(&Bl[0][0]),
                            KB, 64, (u32)K, 4, 3);
        else    tdm_load_2d(Bb + n0, lds_off(&Bl[0][0]),
                            64, KB, (u32)N, 5, 3);
    }
#else
    v4f  areg[8];
    v4f  breg4[4];
    float bregs[16];
#pragma unroll
    for (int i = 0; i < 8; ++i) {
        int idx4 = i * 128 + t;
        int row = idx4 >> 3, c4 = (idx4 & 7) << 2;
        *(v4f*)&Al[0][row * PADA + c4] =
            *(const v4f*)(Ab + (size_t)(mblk + row) * K + c4);
    }
    if (BT) {
#pragma unroll
        for (int i = 0; i < 4; ++i) {
            int idx4 = i * 128 + t;
            int n = idx4 >> 3, c4 = (idx4 & 7) << 2;
            *(v4f*)&Bl[0][n * PADBT + c4] =
                *(const v4f*)(Bb + (size_t)(n0 + n) * K + c4);
        }
    } else {
#pragma unroll
        for (int i = 0; i < 16; ++i) {
            int idx = i * 128 + t;
            int kk = idx >> 6, col = idx & 63;
            Bl[0][kk * PADBN + col] = Bb[(size_t)kk * N + n0 + col];
        }
    }
    __syncthreads();
#endif

    for (int kc = 0; kc < nchunks; ++kc) {
        const int cbuf = kc & 1, nbuf = cbuf ^ 1;
        const int k0n = (kc + 1) * KB;
        const bool more = (kc + 1) < nchunks;

#ifdef USE_TDM
        // ---- prefetch next chunk via TDM, then split-wait on TENSORcnt ----
        if (more) {
            if (w == 0)
                tdm_load_2d(Ab + (size_t)mblk * K + k0n, lds_off(&Al[nbuf][0]),
                            KB, 128, (u32)K, 4, 3);
            if (w == 1) {
                if (BT) tdm_load_2d(Bb + (size_t)n0 * K + k0n, lds_off(&Bl[nbuf][0]),
                                    KB, 64, (u32)K, 4, 3);
                else    tdm_load_2d(Bb + (size_t)k0n * N + n0, lds_off(&Bl[nbuf][0]),
                                    64, KB, (u32)N, 5, 3);
            }
            if (w <= 1) __builtin_amdgcn_s_wait_tensorcnt((short)1);
        } else {
            if (w <= 1) __builtin_amdgcn_s_wait_tensorcnt((short)0);
        }
        __syncthreads();
#else
        if (more) {
#pragma unroll
            for (int i = 0; i < 8; ++i) {
                int idx4 = i * 128 + t;
                int row = idx4 >> 3, c4 = (idx4 & 7) << 2;
                areg[i] = *(const v4f*)(Ab + (size_t)(mblk + row) * K + k0n + c4);
            }
            if (BT) {
#pragma unroll
                for (int i = 0; i < 4; ++i) {
                    int idx4 = i * 128 + t;
                    int n = idx4 >> 3, c4 = (idx4 & 7) << 2;
                    breg4[i] = *(const v4f*)(Bb + (size_t)(n0 + n) * K + k0n + c4);
                }
            } else {
#pragma unroll
                for (int i = 0; i < 16; ++i) {
                    int idx = i * 128 + t;
                    int kk = idx >> 6, col = idx & 63;
                    bregs[i] = Bb[(size_t)(k0n + kk) * N + n0 + col];
                }
            }
        }
#endif

        // ---- compute current chunk from LDS: 8 k-steps x 8 WMMA ----
        const float* Ac0 = &Al[cbuf][(w * 32 + l) * PADA + 2 * h];
        const float* Ac1 = Ac0 + 16 * PADA;
#pragma unroll
        for (int kk = 0; kk < KB; kk += 4) {
            v2f a0 = *(const v2f*)(Ac0 + kk);
            v2f a1 = *(const v2f*)(Ac1 + kk);
#pragma unroll
            for (int s = 0; s < 4; ++s) {
                v2f bv;
                if (BT) {
                    bv = *(const v2f*)&Bl[cbuf][(s * 16 + l) * PADBT + 2 * h + kk];
                } else {
                    const float* bp = &Bl[cbuf][(kk + 2 * h) * PADBN + s * 16 + l];
                    bv.x = bp[0];
                    bv.y = bp[PADBN];
                }
                acc[0][s] = __builtin_amdgcn_wmma_f32_16x16x4_f32(
                    false, a0, false, bv, (short)0, acc[0][s], false, false);
                acc[1][s] = __builtin_amdgcn_wmma_f32_16x16x4_f32(
                    false, a1, false, bv, (short)0, acc[1][s], false, false);
            }
        }

#ifndef USE_TDM
        if (more) {
#pragma unroll
            for (int i = 0; i < 8; ++i) {
                int idx4 = i * 128 + t;
                int row = idx4 >> 3, c4 = (idx4 & 7) << 2;
                *(v4f*)&Al[nbuf][row * PADA + c4] = areg[i];
            }
            if (BT) {
#pragma unroll
                for (int i = 0; i < 4; ++i) {
                    int idx4 = i * 128 + t;
                    int n = idx4 >> 3, c4 = (idx4 & 7) << 2;
                    *(v4f*)&Bl[nbuf][n * PADBT + c4] = breg4[i];
                }
            } else {
#pragma unroll
                for (int i = 0; i < 16; ++i) {
                    int idx = i * 128 + t;
                    int kk = idx >> 6, col = idx & 63;
                    Bl[nbuf][kk * PADBN + col] = bregs[i];
                }
            }
        }
#endif
        __syncthreads();
    }

    // ---- epilogue ----
#pragma unroll
    for (int mi = 0; mi < 2; ++mi)
#pragma unroll
        for (int s = 0; s < 4; ++s)
#pragma unroll
            for (int g = 0; g < 8; ++g) {
                float v = acc[mi][s][g] * scale;
                if (do_relu) v = fmaxf(v, 0.0f);
                Cb[(size_t)(mblk + w * 32 + mi * 16 + g + 8 * h) * N
                   + n0 + s * 16 + l] = v;
            }
}

// ---------------------------------------------------------------------------
// Row sums of att[b,v,:] (one block per row of 512, coalesced).
// ---------------------------------------------------------------------------
__global__ __launch_bounds__(256)
void row_sums(const float* __restrict__ att, float* __restrict__ rowsum)
{
    __shared__ float sh[256];
    const int t = threadIdx.x;
    const float* row = att + (size_t)blockIdx.x * TT;
    sh[t] = row[t] + row[t + 256];
    __syncthreads();
    for (int off = 128; off > 0; off >>= 1) {
        if (t < off) sh[t] += sh[t + off];
        __syncthreads();
    }
    if (t == 0) rowsum[blockIdx.x] = sh[0];
}

// ---------------------------------------------------------------------------
// Column sums of att[b,:,a] (thread owns one column, coalesced each step).
// ---------------------------------------------------------------------------
__global__ __launch_bounds__(256)
void col_sums(const float* __restrict__ att, float* __restrict__ colsum)
{
    const int b = blockIdx.x >> 1;
    const int a = ((blockIdx.x & 1) << 8) + threadIdx.x;
    const float* base = att + (size_t)b * TT * TT + a;
    float acc = 0.0f;
    for (int v = 0; v < TT; ++v)
        acc += base[(size_t)v * TT];
    colsum[b * TT + a] = acc;
}

// ---------------------------------------------------------------------------
// Normalize + threshold with 32x32 LDS tile transpose so BOTH outputs store
// coalesced. Reads raw att in att_v2a's slot, rewrites it in place.
// block = (32,8), grid = (16,16,64)
// ---------------------------------------------------------------------------
__global__ __launch_bounds__(256)
void normalize_att(float* __restrict__ att_v2a, float* __restrict__ att_a2v,
                   const float* __restrict__ rowsum, const float* __restrict__ colsum,
                   const float* __restrict__ thr_p)
{
    __shared__ float tile[32][33];
    const float thr = thr_p[0];
    const int b  = blockIdx.z;
    const int a0 = blockIdx.x * 32;
    const int v0 = blockIdx.y * 32;
    const int tx = threadIdx.x;
    const int ty = threadIdx.y;

    const float cs = colsum[b * TT + a0 + tx];
#pragma unroll
    for (int i = 0; i < 32; i += 8) {
        const int v = v0 + ty + i;
        const size_t idx = ((size_t)b << 18) + ((size_t)v << 9) + a0 + tx;
        const float x = att_v2a[idx];

        float yv = x / (rowsum[b * TT + v] + 1e-8f);
        att_v2a[idx] = (yv > thr) ? yv : 0.0f;

        float ya = x / (cs + 1e-8f);
        tile[ty + i][tx] = (ya > thr) ? ya : 0.0f;
    }
    __syncthreads();
#pragma unroll
    for (int i = 0; i < 32; i += 8) {
        const int a = a0 + ty + i;
        att_a2v[((size_t)b << 18) + ((size_t)a << 9) + v0 + tx] = tile[tx][ty + i];
    }
}

// ---------------------------------------------------------------------------
// Fused residual + LayerNorm (both streams) + average. One block per (b,t).
// ---------------------------------------------------------------------------
__global__ __launch_bounds__(256)
void ln_fuse(const float* __restrict__ v_fea, const float* __restrict__ a_fea,
             const float* __restrict__ cross_v, const float* __restrict__ cross_a,
             const float* __restrict__ gamma, const float* __restrict__ beta,
             float* __restrict__ final_v, float* __restrict__ final_a,
             float* __restrict__ fuse)
{
    __shared__ float sh[4][256];
    const int t = threadIdx.x;
    const size_t row = (size_t)blockIdx.x * DD;

    const float xv = v_fea[row + t] + cross_v[row + t];
    const float xa = a_fea[row + t] + cross_a[row + t];

    sh[0][t] = xv; sh[1][t] = xv * xv;
    sh[2][t] = xa; sh[3][t] = xa * xa;
    __syncthreads();
    for (int off = 128; off > 0; off >>= 1) {
        if (t < off) {
            sh[0][t] += sh[0][t + off];
            sh[1][t] += sh[1][t + off];
            sh[2][t] += sh[2][t + off];
            sh[3][t] += sh[3][t + off];
        }
        __syncthreads();
    }
    const float inv_d = 1.0f / (float)DD;
    const float muv = sh[0][0] * inv_d;
    const float varv = sh[1][0] * inv_d - muv * muv;
    const float mua = sh[2][0] * inv_d;
    const float vara = sh[3][0] * inv_d - mua * mua;

    const float g = gamma[t], bb = beta[t];
    const float fv = (xv - muv) * rsqrtf(varv + 1e-6f) * g + bb;
    const float fa = (xa - mua) * rsqrtf(vara + 1e-6f) * g + bb;

    final_v[row + t] = fv;
    final_a[row + t] = fa;
    fuse[row + t] = (fv + fa) * 0.5f;
}

// ---------------------------------------------------------------------------
extern "C" void kernel_launch(void* const* d_in, const int* in_sizes, int n_in,
                              void* d_out, int out_size, void* d_ws, size_t ws_size,
                              hipStream_t stream)
{
    const float* a_fea = (const float*)d_in[0];
    const float* v_fea = (const float*)d_in[1];
    const float* thr   = (const float*)d_in[2];
    const float* Wv1   = (const float*)d_in[3];
    const float* Wv2   = (const float*)d_in[4];
    const float* Wa1   = (const float*)d_in[5];
    const float* Wa2   = (const float*)d_in[6];
    const float* gamma = (const float*)d_in[7];
    const float* beta  = (const float*)d_in[8];

    float* out = (float*)d_out;
    float* fuse_o    = out;
    float* final_v_o = out + (size_t)BT_ELEMS;
    float* final_a_o = out + (size_t)2 * BT_ELEMS;
    float* cross_v_o = out + (size_t)3 * BT_ELEMS;
    float* cross_a_o = out + (size_t)4 * BT_ELEMS;
    float* att_v2a_o = out + (size_t)5 * BT_ELEMS;
    float* att_a2v_o = att_v2a_o + (size_t)ATT_ELEMS;

    float* ws   = (float*)d_ws;
    float* v_b1 = ws;
    float* v_b2 = ws + (size_t)BT_ELEMS;
    float* a_b1 = ws + (size_t)2 * BT_ELEMS;
    float* a_b2 = ws + (size_t)3 * BT_ELEMS;
    float* rsum = ws + (size_t)4 * BT_ELEMS;
    float* csum = rsum + NB * TT;

    // 1) four ReLU projections: [B*T,256] @ [256,256]
    dim3 gproj(NB * TT / 128, DD / 64, 1);
    gemm_tiled<false><<<gproj, 128, 0, stream>>>(v_fea, Wv1, v_b1, DD, DD, 0, 0, 0, 1.0f, 1);
    gemm_tiled<false><<<gproj, 128, 0, stream>>>(v_fea, Wv2, v_b2, DD, DD, 0, 0, 0, 1.0f, 1);
    gemm_tiled<false><<<gproj, 128, 0, stream>>>(a_fea, Wa1, a_b1, DD, DD, 0, 0, 0, 1.0f, 1);
    gemm_tiled<false><<<gproj, 128, 0, stream>>>(a_fea, Wa2, a_b2, DD, DD, 0, 0, 0, 1.0f, 1);

    // 2) attention scores: att[b] = relu((v_b2[b] @ a_b1[b]^T) / 16)
    //    raw scores parked in att_v2a's output slot (rewritten below)
    dim3 gatt(TT / 128, TT / 64, NB);
    gemm_tiled<true><<<gatt, 128, 0, stream>>>(v_b2, a_b1, att_v2a_o, DD, TT,
                                               (long)TT * DD, (long)TT * DD,
                                               (long)TT * TT, 1.0f / 16.0f, 1);

    // 3) row / column sums, then normalize + threshold (+ tiled transpose)
    row_sums<<<NB * TT, 256, 0, stream>>>(att_v2a_o, rsum);
    col_sums<<<NB * 2, 256, 0, stream>>>(att_v2a_o, csum);
    normalize_att<<<dim3(TT / 32, TT / 32, NB), dim3(32, 8), 0, stream>>>(
        att_v2a_o, att_a2v_o, rsum, csum, thr);

    // 4) cross features: [512,512] @ [512,256] per batch
    dim3 gcross(TT / 128, DD / 64, NB);
    gemm_tiled<false><<<gcross, 128, 0, stream>>>(att_v2a_o, a_b2, cross_v_o, TT, DD,
                                                  (long)TT * TT, (long)TT * DD,
                                                  (long)TT * DD, 1.0f, 0);
    gemm_tiled<false><<<gcross, 128, 0, stream>>>(att_a2v_o, v_b1, cross_a_o, TT, DD,
                                                  (long)TT * TT, (long)TT * DD,
                                                  (long)TT * DD, 1.0f, 0);

    // 5) residual + LayerNorm + average
    ln_fuse<<<NB * TT, 256, 0, stream>>>(v_fea, a_fea, cross_v_o, cross_a_o,
                                         gamma, beta, final_v_o, final_a_o,
                                         fuse_o);
}